// GNNOntologyLayer_1382979470011
// MI455X (gfx1250) — compile-verified
//
#include <hip/hip_runtime.h>
#include <math.h>

typedef __attribute__((ext_vector_type(2))) float v2f;
typedef __attribute__((ext_vector_type(8))) float v8f;

#define NB      2
#define NNODES  20000
#define NEDGES  320000
#define DIM     128
#define ROWS    (NB*NNODES)      /* 40000, divisible by 16 */
#define QKSCALE 0.25f            /* 1/sqrt(HEAD_DIM=16)    */

// ---------------------------------------------------------------- init
__global__ void k_init(float* __restrict__ out, float* __restrict__ m,
                       float* __restrict__ s, float* __restrict__ counts) {
    int i = blockIdx.x * blockDim.x + threadIdx.x;
    if (i < ROWS * DIM) out[i] = 0.0f;
    if (i < ROWS)       { m[i] = -3.0e38f; s[i] = 0.0f; }
    if (i < NNODES)     counts[i] = 0.0f;
}

// ------------------------------------------------- GEMM: H = X @ Wproj
// block = 256 (8 waves); block handles 16 rows x 128 cols; wave = 16x16 tile
__global__ void __launch_bounds__(256)
k_gemm_h(const float* __restrict__ X, const float* __restrict__ W,
         float* __restrict__ H) {
    const int lane = threadIdx.x & 31;
    const int wv   = threadIdx.x >> 5;
    const int m0   = blockIdx.x * 16;
    const int n0   = wv * 16;
    const int half = lane >> 4;
    const int l    = lane & 15;
    const float* __restrict__ Arow = X + (size_t)(m0 + l) * DIM;
    v8f c = {0.f,0.f,0.f,0.f,0.f,0.f,0.f,0.f};
#pragma unroll
    for (int k0 = 0; k0 < DIM; k0 += 4) {
        const int ka = k0 + half * 2;
        v2f a, b;
        a.x = Arow[ka];
        a.y = Arow[ka + 1];
        b.x = W[(size_t)ka * DIM + n0 + l];
        b.y = W[(size_t)(ka + 1) * DIM + n0 + l];
        c = __builtin_amdgcn_wmma_f32_16x16x4_f32(false, a, false, b,
                                                  (short)0, c, false, false);
    }
#pragma unroll
    for (int r = 0; r < 8; ++r) {
        const int row = m0 + r + half * 8;
        H[(size_t)row * DIM + n0 + l] = c[r];
    }
}

// --------------- fused node-level QKV: stores Vn and qk = dot(Q,K)*scale
__global__ void __launch_bounds__(256)
k_gemm_qkv(const float* __restrict__ H,
           const float* __restrict__ Wq, const float* __restrict__ bq,
           const float* __restrict__ Wk, const float* __restrict__ bk,
           const float* __restrict__ Wv, const float* __restrict__ bv,
           float* __restrict__ Vout, float* __restrict__ qk) {
    __shared__ float qkLds[16];
    const int lane = threadIdx.x & 31;
    const int wv   = threadIdx.x >> 5;
    const int m0   = blockIdx.x * 16;
    const int n0   = wv * 16;
    const int half = lane >> 4;
    const int l    = lane & 15;
    if (threadIdx.x < 16) qkLds[threadIdx.x] = 0.0f;
    __syncthreads();

    const float* __restrict__ Arow = H + (size_t)(m0 + l) * DIM;
    v8f cq = {0.f,0.f,0.f,0.f,0.f,0.f,0.f,0.f};
    v8f ck = {0.f,0.f,0.f,0.f,0.f,0.f,0.f,0.f};
    v8f cv = {0.f,0.f,0.f,0.f,0.f,0.f,0.f,0.f};
#pragma unroll
    for (int k0 = 0; k0 < DIM; k0 += 4) {
        const int ka = k0 + half * 2;
        v2f a, b;
        a.x = Arow[ka];
        a.y = Arow[ka + 1];
        b.x = Wq[(size_t)ka * DIM + n0 + l];
        b.y = Wq[(size_t)(ka + 1) * DIM + n0 + l];
        cq = __builtin_amdgcn_wmma_f32_16x16x4_f32(false, a, false, b,
                                                   (short)0, cq, false, false);
        b.x = Wk[(size_t)ka * DIM + n0 + l];
        b.y = Wk[(size_t)(ka + 1) * DIM + n0 + l];
        ck = __builtin_amdgcn_wmma_f32_16x16x4_f32(false, a, false, b,
                                                   (short)0, ck, false, false);
        b.x = Wv[(size_t)ka * DIM + n0 + l];
        b.y = Wv[(size_t)(ka + 1) * DIM + n0 + l];
        cv = __builtin_amdgcn_wmma_f32_16x16x4_f32(false, a, false, b,
                                                   (short)0, cv, false, false);
    }
    const float addq = bq[n0 + l], addk = bk[n0 + l], addv = bv[n0 + l];
#pragma unroll
    for (int r = 0; r < 8; ++r) {
        const float qv = cq[r] + addq;
        const float kv = ck[r] + addk;
        const float vv = cv[r] + addv;
        const int row  = m0 + r + half * 8;
        Vout[(size_t)row * DIM + n0 + l] = vv;
        // reduce q*k over the 16 lanes (N columns) of this half-wave
        float p = qv * kv;
        p += __shfl_xor(p, 1);
        p += __shfl_xor(p, 2);
        p += __shfl_xor(p, 4);
        p += __shfl_xor(p, 8);
        if (l == 0) atomicAdd(&qkLds[r + half * 8], p);
    }
    __syncthreads();
    if (threadIdx.x < 16)
        qk[m0 + threadIdx.x] = qkLds[threadIdx.x] * QKSCALE;
}

// ------------------------------------------------------- degree counts
__global__ void k_count(const int* __restrict__ dst, float* __restrict__ counts) {
    int e = blockIdx.x * blockDim.x + threadIdx.x;
    if (e < NEDGES) atomicAdd(&counts[dst[e]], 1.0f);
}
__global__ void k_invc(const float* __restrict__ counts, float* __restrict__ invc) {
    int n = blockIdx.x * blockDim.x + threadIdx.x;
    if (n < NNODES) invc[n] = 1.0f / fmaxf(counts[n], 1.0f);
}

// ------------------------------------------ segment max (float via int)
__global__ void k_segmax(const int* __restrict__ src, const int* __restrict__ dst,
                         const float* __restrict__ qk, float* __restrict__ m) {
    int i = blockIdx.x * blockDim.x + threadIdx.x;
    if (i >= NB * NEDGES) return;
    const int b = i / NEDGES, e = i - b * NEDGES;
    const float a = qk[b * NNODES + src[e]];
    float* addr = &m[b * NNODES + dst[e]];
    if (a >= 0.0f) atomicMax((int*)addr, __float_as_int(a));
    else           atomicMin((unsigned int*)addr, __float_as_uint(a));
}

// --------------------------------------- exp(a - m[dst]) and segment sum
__global__ void k_exps(const int* __restrict__ src, const int* __restrict__ dst,
                       const float* __restrict__ qk, const float* __restrict__ m,
                       float* __restrict__ s, float* __restrict__ ebuf) {
    int i = blockIdx.x * blockDim.x + threadIdx.x;
    if (i >= NB * NEDGES) return;
    const int b = i / NEDGES, e = i - b * NEDGES;
    const int dn = dst[e];
    const float a  = qk[b * NNODES + src[e]];
    const float ee = __expf(a - m[b * NNODES + dn]);
    ebuf[i] = ee;
    atomicAdd(&s[b * NNODES + dn], ee);
}

// ------------- fused scatter: out[dst] += w*V[src] + invCount[dst]*h[src]
// one wave per (batch, edge); lane handles 4 channels
__global__ void __launch_bounds__(256)
k_scatter(const int* __restrict__ src, const int* __restrict__ dst,
          const float* __restrict__ ebuf, const float* __restrict__ s,
          const float* __restrict__ invc, const float* __restrict__ Vn,
          const float* __restrict__ H, float* __restrict__ out) {
    const int gw   = (int)((blockIdx.x * blockDim.x + threadIdx.x) >> 5);
    const int lane = threadIdx.x & 31;
    if (gw >= NB * NEDGES) return;
    const int b = gw / NEDGES, e = gw - b * NEDGES;
    const int sn = src[e], dn = dst[e];
    const size_t srow = (size_t)(b * NNODES + sn) * DIM;
    const size_t drow = (size_t)(b * NNODES + dn) * DIM;
    const float w  = ebuf[gw] / s[b * NNODES + dn];
    const float iv = invc[dn];
    const int c0 = lane * 4;
    const float4 v  = *(const float4*)(Vn + srow + c0);
    const float4 h4 = *(const float4*)(H  + srow + c0);
    atomicAdd(out + drow + c0 + 0, w * v.x + iv * h4.x);
    atomicAdd(out + drow + c0 + 1, w * v.y + iv * h4.y);
    atomicAdd(out + drow + c0 + 2, w * v.z + iv * h4.z);
    atomicAdd(out + drow + c0 + 3, w * v.w + iv * h4.w);
}

// --------------------------------------------- LayerNorm, wave per row
__global__ void __launch_bounds__(256)
k_ln(float* __restrict__ out, const float* __restrict__ g,
     const float* __restrict__ bb) {
    const int gw   = (int)((blockIdx.x * blockDim.x + threadIdx.x) >> 5);
    const int lane = threadIdx.x & 31;
    if (gw >= ROWS) return;
    const size_t row = (size_t)gw * DIM;
    const int c0 = lane * 4;
    float4 v = *(const float4*)(out + row + c0);
    float sum = v.x + v.y + v.z + v.w;
    float sq  = v.x*v.x + v.y*v.y + v.z*v.z + v.w*v.w;
#pragma unroll
    for (int o = 16; o >= 1; o >>= 1) {
        sum += __shfl_xor(sum, o);
        sq  += __shfl_xor(sq,  o);
    }
    const float mu  = sum * (1.0f / 128.0f);
    const float var = sq * (1.0f / 128.0f) - mu * mu;
    const float rs  = rsqrtf(var + 1e-5f);
    float4 o4;
    o4.x = (v.x - mu) * rs * g[c0 + 0] + bb[c0 + 0];
    o4.y = (v.y - mu) * rs * g[c0 + 1] + bb[c0 + 1];
    o4.z = (v.z - mu) * rs * g[c0 + 2] + bb[c0 + 2];
    o4.w = (v.w - mu) * rs * g[c0 + 3] + bb[c0 + 3];
    *(float4*)(out + row + c0) = o4;
}

// -------------------------------------------------------------- driver
extern "C" void kernel_launch(void* const* d_in, const int* in_sizes, int n_in,
                              void* d_out, int out_size, void* d_ws, size_t ws_size,
                              hipStream_t stream) {
    const float* x      = (const float*)d_in[0];
    const float* proj_w = (const float*)d_in[1];
    const float* wq_w   = (const float*)d_in[2];
    const float* wq_b   = (const float*)d_in[3];
    const float* wk_w   = (const float*)d_in[4];
    const float* wk_b   = (const float*)d_in[5];
    const float* wv_w   = (const float*)d_in[6];
    const float* wv_b   = (const float*)d_in[7];
    const float* ln_g   = (const float*)d_in[8];
    const float* ln_b   = (const float*)d_in[9];
    const int*   ei     = (const int*)d_in[10];
    const int* src = ei;
    const int* dst = ei + NEDGES;
    float* out = (float*)d_out;

    float* ws     = (float*)d_ws;
    float* H      = ws;                              // ROWS*DIM
    float* Vn     = H  + (size_t)ROWS * DIM;         // ROWS*DIM
    float* qk     = Vn + (size_t)ROWS * DIM;         // ROWS
    float* mbuf   = qk + ROWS;                       // ROWS
    float* sbuf   = mbuf + ROWS;                     // ROWS
    float* counts = sbuf + ROWS;                     // NNODES
    float* invc   = counts + NNODES;                 // NNODES
    float* ebuf   = invc + NNODES;                   // NB*NEDGES

    k_init   <<<(ROWS * DIM + 255) / 256, 256, 0, stream>>>(out, mbuf, sbuf, counts);
    k_gemm_h <<<ROWS / 16, 256, 0, stream>>>(x, proj_w, H);
    k_gemm_qkv<<<ROWS / 16, 256, 0, stream>>>(H, wq_w, wq_b, wk_w, wk_b,
                                              wv_w, wv_b, Vn, qk);
    k_count  <<<(NEDGES + 255) / 256, 256, 0, stream>>>(dst, counts);
    k_invc   <<<(NNODES + 255) / 256, 256, 0, stream>>>(counts, invc);
    k_segmax <<<(NB * NEDGES + 255) / 256, 256, 0, stream>>>(src, dst, qk, mbuf);
    k_exps   <<<(NB * NEDGES + 255) / 256, 256, 0, stream>>>(src, dst, qk, mbuf, sbuf, ebuf);
    k_scatter<<<((size_t)NB * NEDGES * 32 + 255) / 256, 256, 0, stream>>>(
        src, dst, ebuf, sbuf, invc, Vn, H, out);
    k_ln     <<<((size_t)ROWS * 32 + 255) / 256, 256, 0, stream>>>(out, ln_g, ln_b);
}